// GraphSAGE_9998683865369
// MI455X (gfx1250) — compile-verified
//
#include <hip/hip_runtime.h>

// ---------------------------------------------------------------------------
// GraphSAGE 2-layer forward for MI455X (gfx1250, wave32).
//  - Edge phase: 1 wave per edge, float4 gather + global_atomic_add_f32 scatter.
//    agg (25.6MB) is L2-resident (192MB L2), atomics RMW in L2.
//  - Node phase: fp32 WMMA (V_WMMA_F32_16X16X4_F32), one wave per 16x16 output
//    tile, fused mean-scale + dual-GEMM (W_l, W_r) + bias + ReLU epilogue.
// ---------------------------------------------------------------------------

typedef __attribute__((ext_vector_type(2))) float v2f;
typedef __attribute__((ext_vector_type(8))) float v8f;

#define IN_DIM  128
#define HID_DIM 128
#define OUT_DIM 64

// ------------------------- degree accumulation -----------------------------
__global__ void degree_kernel(const int* __restrict__ dst,
                              float* __restrict__ deg, int E) {
  int e = blockIdx.x * blockDim.x + threadIdx.x;
  if (e < E) atomicAdd(&deg[dst[e]], 1.0f);
}

// deg -> deg_inv in place (deg >= 1 whenever > 0, so 1/deg matches reference)
__global__ void deg_inv_kernel(float* __restrict__ deg, int N) {
  int i = blockIdx.x * blockDim.x + threadIdx.x;
  if (i < N) {
    float d = deg[i];
    deg[i] = (d > 0.0f) ? (1.0f / d) : 0.0f;
  }
}

// --------------------- edge scatter: agg[dst] += feat[src] -----------------
// One wave32 per edge; lane l owns features [4l, 4l+4). src/dst are wave-
// uniform (scalar loads), gather is a coalesced 512B float4 burst per wave.
__global__ __launch_bounds__(256)
void scatter_add_kernel(const float* __restrict__ feat,   // [N,128]
                        const int* __restrict__ src,
                        const int* __restrict__ dst,
                        float* __restrict__ agg,          // [N,128]
                        int E) {
  int wid = blockIdx.x * (blockDim.x >> 5) + (threadIdx.x >> 5);
  if (wid >= E) return;
  int lane = threadIdx.x & 31;
  int s = src[wid];
  int d = dst[wid];
  float4 v = ((const float4*)(feat + (size_t)s * 128))[lane];
  float* o = agg + (size_t)d * 128 + lane * 4;
  atomicAdd(o + 0, v.x);
  atomicAdd(o + 1, v.y);
  atomicAdd(o + 2, v.z);
  atomicAdd(o + 3, v.w);
}

// ----------------- fused SAGE node update via fp32 WMMA --------------------
// out[r, c] = act( (agg[r]*deg_inv[r]) @ Wl + xin @ Wr + bias )
// One wave computes a 16x16 tile of out. K and M are multiples of 4 / 16.
//
// V_WMMA_F32_16X16X4_F32 operand layouts (ISA 7.12.2, wave32):
//   A (16x4): lane&15 = row M; lane>=16 holds K={k+2,k+3}, else K={k,k+1}
//   B (4x16): lane&15 = col N; VGPR0/1 = rows {k,k+1} (lanes<16) or {k+2,k+3}
//   C/D v8f : VGPR r = row r (lanes<16) or row r+8 (lanes>=16), col = lane&15
template <int K, int M, bool RELU>
__global__ __launch_bounds__(256)
void sage_gemm_kernel(const float* __restrict__ agg,      // [N,K] neighbor sums
                      const float* __restrict__ xin,      // [N,K] root feats
                      const float* __restrict__ deg_inv,  // [N]
                      const float* __restrict__ Wl,       // [K,M]
                      const float* __restrict__ Wr,       // [K,M]
                      const float* __restrict__ bias,     // [M]
                      float* __restrict__ out,            // [N,M]
                      int n_tiles) {
  int wid = blockIdx.x * (blockDim.x >> 5) + (threadIdx.x >> 5);
  if (wid >= n_tiles) return;                 // whole-wave uniform: EXEC stays all-1s

  constexpr int COL_TILES = M / 16;
  const int row0 = (wid / COL_TILES) * 16;
  const int col0 = (wid % COL_TILES) * 16;
  const int lane = threadIdx.x & 31;
  const int half = lane >> 4;                 // 0: K={k,k+1}, 1: K={k+2,k+3}
  const int l16  = lane & 15;

  const float* arow = agg + (size_t)(row0 + l16) * K + 2 * half;
  const float* xrow = xin + (size_t)(row0 + l16) * K + 2 * half;
  const float* wl   = Wl + (size_t)(2 * half) * M + col0 + l16;
  const float* wr   = Wr + (size_t)(2 * half) * M + col0 + l16;
  const float dinv  = deg_inv[row0 + l16];

  v8f acc = {};
#pragma unroll 4
  for (int k = 0; k < K; k += 4) {
    // GEMM 1: mean aggregation times Wl (fold deg_inv into the A fragment)
    v2f a = *(const v2f*)(arow + k);
    a.x *= dinv;
    a.y *= dinv;
    v2f bl;
    bl.x = wl[(size_t)(k + 0) * M];
    bl.y = wl[(size_t)(k + 1) * M];
    acc = __builtin_amdgcn_wmma_f32_16x16x4_f32(false, a, false, bl,
                                                (short)0, acc, false, false);
    // GEMM 2: root features times Wr, same accumulator
    v2f ax = *(const v2f*)(xrow + k);
    v2f br;
    br.x = wr[(size_t)(k + 0) * M];
    br.y = wr[(size_t)(k + 1) * M];
    acc = __builtin_amdgcn_wmma_f32_16x16x4_f32(false, ax, false, br,
                                                (short)0, acc, false, false);
  }

  const float bv = bias[col0 + l16];
  float* orow = out + (size_t)(row0 + 8 * half) * M + col0 + l16;
#pragma unroll
  for (int r = 0; r < 8; ++r) {
    float v = acc[r] + bv;
    if (RELU) v = fmaxf(v, 0.0f);
    orow[(size_t)r * M] = v;
  }
}

// ---------------------------------------------------------------------------
extern "C" void kernel_launch(void* const* d_in, const int* in_sizes, int n_in,
                              void* d_out, int out_size, void* d_ws, size_t ws_size,
                              hipStream_t stream) {
  const float* x   = (const float*)d_in[0];   // [N,128]
  const int*   ei  = (const int*)d_in[1];     // [2,E] (int32 per harness)
  const float* W1l = (const float*)d_in[2];   // [128,128]
  const float* b1  = (const float*)d_in[3];   // [128]
  const float* W1r = (const float*)d_in[4];   // [128,128]
  const float* W2l = (const float*)d_in[5];   // [128,64]
  const float* b2  = (const float*)d_in[6];   // [64]
  const float* W2r = (const float*)d_in[7];   // [128,64]
  float* out = (float*)d_out;                 // [N,64]

  const int N = in_sizes[0] / IN_DIM;         // 50000 (multiple of 16)
  const int E = in_sizes[1] / 2;              // 800000
  const int* src = ei;
  const int* dst = ei + E;

  // Workspace: deg/deg_inv (N) | agg (N*128, reused per layer) | h (N*128)
  float* ws  = (float*)d_ws;
  float* deg = ws;
  float* agg = ws + (size_t)((N + 3) & ~3);
  float* h   = agg + (size_t)N * HID_DIM;

  hipMemsetAsync(deg, 0, (size_t)N * sizeof(float), stream);
  hipMemsetAsync(agg, 0, (size_t)N * HID_DIM * sizeof(float), stream);

  degree_kernel<<<(E + 255) / 256, 256, 0, stream>>>(dst, deg, E);
  deg_inv_kernel<<<(N + 255) / 256, 256, 0, stream>>>(deg, N);

  // ---- Layer 1 ----
  scatter_add_kernel<<<(E + 7) / 8, 256, 0, stream>>>(x, src, dst, agg, E);
  {
    const int tiles = (N / 16) * (HID_DIM / 16);
    sage_gemm_kernel<IN_DIM, HID_DIM, true>
        <<<(tiles + 7) / 8, 256, 0, stream>>>(agg, x, deg, W1l, W1r, b1, h, tiles);
  }

  // ---- Layer 2 ----
  hipMemsetAsync(agg, 0, (size_t)N * HID_DIM * sizeof(float), stream);
  scatter_add_kernel<<<(E + 7) / 8, 256, 0, stream>>>(h, src, dst, agg, E);
  {
    const int tiles = (N / 16) * (OUT_DIM / 16);
    sage_gemm_kernel<HID_DIM, OUT_DIM, false>
        <<<(tiles + 7) / 8, 256, 0, stream>>>(agg, h, deg, W2l, W2r, b2, out, tiles);
  }
}